// GlobalFusion_75445395522211
// MI455X (gfx1250) — compile-verified
//
#include <hip/hip_runtime.h>
#include <hip/hip_bf16.h>
#include <stdint.h>

#define DIM    96
#define DIM3   (DIM*DIM*DIM)      /* 884736 */
#define CH     24
#define N_CUR  150000
#define N_GLB  300000
#define N_TGT  100000
#define N_GTGT 200000

typedef float v4f __attribute__((ext_vector_type(4)));

__device__ __forceinline__ void wait_async0() {
#if defined(__has_builtin)
#if __has_builtin(__builtin_amdgcn_s_wait_asynccnt)
  __builtin_amdgcn_s_wait_asynccnt(0);
#else
  asm volatile("s_wait_asynccnt 0" ::: "memory");
#endif
#else
  asm volatile("s_wait_asynccnt 0" ::: "memory");
#endif
}

// ---------------------------------------------------------------------------
// Streaming fill via CDNA5 async LDS->global stores (ASYNCcnt path).
// Each lane owns a 16B LDS slot holding the 32-bit pattern; the loop issues
// global_store_async_from_lds_b128 in GVS mode (saddr = base, vaddr = 32-bit
// byte offset). All fill regions are < 2^31 bytes, so the loop is pure 32-bit
// arithmetic. Hardware caps outstanding async ops at 63 per wave.
// ---------------------------------------------------------------------------
__global__ __launch_bounds__(256) void k_fill_async(unsigned* __restrict__ base,
                                                    unsigned n16, unsigned pattern) {
  __shared__ unsigned buf[1024];              // 4 KB at LDS offset 0
  for (int i = threadIdx.x; i < 1024; i += blockDim.x) buf[i] = pattern;
  __syncthreads();
  unsigned lds_addr = (unsigned)threadIdx.x * 16u;
  unsigned stride   = gridDim.x * blockDim.x * 16u;           // bytes per step
  unsigned off      = (blockIdx.x * blockDim.x + threadIdx.x) * 16u;
  unsigned end      = n16 * 16u;
  for (; off < end; off += stride) {
    asm volatile("global_store_async_from_lds_b128 %0, %1, %2"
                 :: "v"(off), "v"(lds_addr), "s"(base)
                 : "memory");
  }
  wait_async0();
}

__global__ __launch_bounds__(256) void k_scatter_cur(const int* __restrict__ cc,
                                                     const unsigned char* __restrict__ gmask,
                                                     const unsigned char* __restrict__ occ,
                                                     int* __restrict__ claim_cur,
                                                     unsigned char* __restrict__ valid_vol,
                                                     unsigned char* __restrict__ occ_vol) {
  int i = blockIdx.x * blockDim.x + threadIdx.x;
  if (i >= N_CUR) return;
  int x = cc[3 * i], y = cc[3 * i + 1], z = cc[3 * i + 2];
  int v = (x * DIM + y) * DIM + z;
  if (gmask[i]) valid_vol[v] = 1;     // all writers store 1: race-free
  if (occ[i])   occ_vol[v]   = 1;
  atomicMax(&claim_cur[v], i);        // last index wins deterministically
}

__global__ __launch_bounds__(256) void k_glb(const int* __restrict__ gc,
                                             const int* __restrict__ origin,
                                             const unsigned char* __restrict__ valid_vol,
                                             int* __restrict__ claim_glb,
                                             float* __restrict__ out_valid,
                                             float* __restrict__ out_near) {
  int i = blockIdx.x * blockDim.x + threadIdx.x;
  if (i >= N_GLB) return;
  int x = gc[3 * i]     - origin[0];
  int y = gc[3 * i + 1] - origin[1];
  int z = gc[3 * i + 2] - origin[2];
  bool inb = (x >= 0) & (x < DIM) & (y >= 0) & (y < DIM) & (z >= 0) & (z < DIM);
  int xc = min(max(x, 0), DIM - 1), yc = min(max(y, 0), DIM - 1), zc = min(max(z, 0), DIM - 1);
  bool valid = inb && (valid_vol[(xc * DIM + yc) * DIM + zc] != 0);
  out_valid[i] = valid ? 1.0f : 0.0f;
  out_near[i]  = (inb && !valid) ? 1.0f : 0.0f;
  if (valid) atomicMax(&claim_glb[(x * DIM + y) * DIM + z], i);
}

// One kernel for both target claim sets: current-fragment targets get claim
// value N_GTGT + j, so under atomicMax they always beat global targets
// (matching the "current scattered second / later write wins" semantics).
__global__ __launch_bounds__(256) void k_tgt_claim(const int* __restrict__ gtc,
                                                   const int* __restrict__ ctc,
                                                   const int* __restrict__ origin,
                                                   int* __restrict__ claim_tgt,
                                                   float* __restrict__ out_vtgt) {
  int i = blockIdx.x * blockDim.x + threadIdx.x;
  if (i < N_GTGT) {
    int x = gtc[3 * i]     - origin[0];
    int y = gtc[3 * i + 1] - origin[1];
    int z = gtc[3 * i + 2] - origin[2];
    bool vt = (x >= 0) & (x < DIM) & (y >= 0) & (y < DIM) & (z >= 0) & (z < DIM);
    out_vtgt[i] = vt ? 1.0f : 0.0f;
    if (vt) atomicMax(&claim_tgt[(x * DIM + y) * DIM + z], i);
  } else if (i < N_GTGT + N_TGT) {
    int j = i - N_GTGT;
    int x = ctc[3 * j], y = ctc[3 * j + 1], z = ctc[3 * j + 2];
    atomicMax(&claim_tgt[(x * DIM + y) * DIM + z], N_GTGT + j);
  }
}

__global__ __launch_bounds__(256) void k_write_cur(const int* __restrict__ cc,
                                                   const float* __restrict__ vals,
                                                   const int* __restrict__ claim_cur,
                                                   float* __restrict__ out_curv) {
  int i = blockIdx.x * blockDim.x + threadIdx.x;
  if (i >= N_CUR) return;
  int x = cc[3 * i], y = cc[3 * i + 1], z = cc[3 * i + 2];
  int v = (x * DIM + y) * DIM + z;
  if (claim_cur[v] != i) return;
  const v4f* src = (const v4f*)(vals + (size_t)i * CH);     // 96B rows: 16B aligned
  v4f* dst = (v4f*)(out_curv + (size_t)v * CH);
#pragma unroll
  for (int k = 0; k < CH / 4; ++k)
    __builtin_nontemporal_store(__builtin_nontemporal_load(&src[k]), &dst[k]);
}

__global__ __launch_bounds__(256) void k_write_glb(const int* __restrict__ gc,
                                                   const float* __restrict__ vals,
                                                   const int* __restrict__ origin,
                                                   const int* __restrict__ claim_glb,
                                                   const float* __restrict__ out_valid,
                                                   float* __restrict__ out_glbv,
                                                   unsigned char* __restrict__ upd_flag) {
  int i = blockIdx.x * blockDim.x + threadIdx.x;
  if (i >= N_GLB) return;
  if (out_valid[i] == 0.0f) return;                    // computed by k_glb
  int x = gc[3 * i]     - origin[0];
  int y = gc[3 * i + 1] - origin[1];
  int z = gc[3 * i + 2] - origin[2];
  int v = (x * DIM + y) * DIM + z;
  if (claim_glb[v] != i) return;
  const v4f* src = (const v4f*)(vals + (size_t)i * CH);
  v4f* dst = (v4f*)(out_glbv + (size_t)v * CH);
  bool anynz = false;
#pragma unroll
  for (int k = 0; k < CH / 4; ++k) {
    v4f r = __builtin_nontemporal_load(&src[k]);
    anynz |= (r.x != 0.0f) | (r.y != 0.0f) | (r.z != 0.0f) | (r.w != 0.0f);
    __builtin_nontemporal_store(r, &dst[k]);
  }
  if (anynz) upd_flag[v] = 1;      // fused (global_volume != 0).any(-1)
}

__global__ __launch_bounds__(256) void k_write_tgt(const int* __restrict__ gtc,
                                                   const int* __restrict__ ctc,
                                                   const int* __restrict__ origin,
                                                   const float* __restrict__ gtsdf,
                                                   const float* __restrict__ ctsdf,
                                                   const int* __restrict__ claim_tgt,
                                                   float* __restrict__ out_tgtv) {
  int i = blockIdx.x * blockDim.x + threadIdx.x;
  if (i < N_GTGT) {
    int x = gtc[3 * i]     - origin[0];
    int y = gtc[3 * i + 1] - origin[1];
    int z = gtc[3 * i + 2] - origin[2];
    bool vt = (x >= 0) & (x < DIM) & (y >= 0) & (y < DIM) & (z >= 0) & (z < DIM);
    if (!vt) return;
    int v = (x * DIM + y) * DIM + z;
    if (claim_tgt[v] == i) out_tgtv[v] = gtsdf[i];
  } else if (i < N_GTGT + N_TGT) {
    int j = i - N_GTGT;
    int x = ctc[3 * j], y = ctc[3 * j + 1], z = ctc[3 * j + 2];
    int v = (x * DIM + y) * DIM + z;
    if (claim_tgt[v] == N_GTGT + j) out_tgtv[v] = ctsdf[j];
  }
}

// 4 voxels/thread: packed uint reads of both byte volumes, one NT v4f store.
__global__ __launch_bounds__(256) void k_mask(const unsigned* __restrict__ flag4,
                                              const unsigned* __restrict__ occ4,
                                              float* __restrict__ out_upd) {
  int i = blockIdx.x * blockDim.x + threadIdx.x;
  if (i >= DIM3 / 4) return;
  unsigned f = flag4[i] | occ4[i];
  v4f r;
  r.x = (f & 0x000000FFu) ? 1.0f : 0.0f;
  r.y = (f & 0x0000FF00u) ? 1.0f : 0.0f;
  r.z = (f & 0x00FF0000u) ? 1.0f : 0.0f;
  r.w = (f & 0xFF000000u) ? 1.0f : 0.0f;
  __builtin_nontemporal_store(r, (v4f*)out_upd + i);
}

extern "C" void kernel_launch(void* const* d_in, const int* in_sizes, int n_in,
                              void* d_out, int out_size, void* d_ws, size_t ws_size,
                              hipStream_t stream) {
  const int*           cur_coords  = (const int*)d_in[0];
  const float*         cur_vals    = (const float*)d_in[1];
  const int*           glb_coords  = (const int*)d_in[2];
  const float*         glb_vals    = (const float*)d_in[3];
  const int*           tgt_coords  = (const int*)d_in[4];
  const float*         tsdf_tgt    = (const float*)d_in[5];
  const int*           gtgt_coords = (const int*)d_in[6];
  const float*         gtsdf_tgt   = (const float*)d_in[7];
  const int*           origin      = (const int*)d_in[8];
  const unsigned char* gmask       = (const unsigned char*)d_in[9];
  const unsigned char* occ         = (const unsigned char*)d_in[10];

  float* out       = (float*)d_out;
  float* out_upd   = out;
  float* out_curv  = out_upd  + DIM3;
  float* out_glbv  = out_curv + (size_t)DIM3 * CH;
  float* out_tgtv  = out_glbv + (size_t)DIM3 * CH;
  float* out_valid = out_tgtv + DIM3;
  float* out_vtgt  = out_valid + N_GLB;
  float* out_near  = out_vtgt  + N_GTGT;

  // ws layout: claim_cur | claim_glb | claim_tgt (int32, DIM3 each),
  // then valid_vol | occ_vol | upd_flag (uint8, DIM3 each). ~13.3 MB total.
  int* claim_cur = (int*)d_ws;
  int* claim_glb = claim_cur + DIM3;
  int* claim_tgt = claim_glb + DIM3;
  unsigned char* valid_vol = (unsigned char*)(claim_tgt + DIM3);
  unsigned char* occ_vol   = valid_vol + DIM3;
  unsigned char* upd_flag  = occ_vol + DIM3;

  const int T = 256;
  dim3 fillGrid(2048);

  // Phase 1: async-stream fills.
  //  - current_volume + global_volume are adjacent zeros: one fill (~170 MB)
  //  - target_volume: 1.0f pattern
  //  - claim volumes: 0xFFFFFFFF (== -1), adjacent: one fill
  //  - byte volumes (valid/occ/flag): 0, adjacent: one fill
  k_fill_async<<<fillGrid, T, 0, stream>>>((unsigned*)out_curv,
                                           (unsigned)(2u * DIM3 * CH / 4), 0u);
  k_fill_async<<<fillGrid, T, 0, stream>>>((unsigned*)out_tgtv,
                                           (unsigned)(DIM3 / 4), 0x3F800000u);
  k_fill_async<<<fillGrid, T, 0, stream>>>((unsigned*)claim_cur,
                                           (unsigned)(3u * DIM3 / 4), 0xFFFFFFFFu);
  k_fill_async<<<fillGrid, T, 0, stream>>>((unsigned*)valid_vol,
                                           (unsigned)(3u * DIM3 / 16), 0u);

  // Phase 2: scatters / claims.
  k_scatter_cur<<<(N_CUR + T - 1) / T, T, 0, stream>>>(cur_coords, gmask, occ,
                                                       claim_cur, valid_vol, occ_vol);
  k_glb<<<(N_GLB + T - 1) / T, T, 0, stream>>>(glb_coords, origin, valid_vol,
                                               claim_glb, out_valid, out_near);
  k_tgt_claim<<<(N_GTGT + N_TGT + T - 1) / T, T, 0, stream>>>(gtgt_coords, tgt_coords,
                                                              origin, claim_tgt, out_vtgt);

  // Phase 3: winner write-back.
  k_write_cur<<<(N_CUR + T - 1) / T, T, 0, stream>>>(cur_coords, cur_vals, claim_cur, out_curv);
  k_write_glb<<<(N_GLB + T - 1) / T, T, 0, stream>>>(glb_coords, glb_vals, origin,
                                                     claim_glb, out_valid, out_glbv, upd_flag);
  k_write_tgt<<<(N_GTGT + N_TGT + T - 1) / T, T, 0, stream>>>(gtgt_coords, tgt_coords, origin,
                                                              gtsdf_tgt, tsdf_tgt,
                                                              claim_tgt, out_tgtv);

  // Phase 4: updated_mask.
  k_mask<<<(DIM3 / 4 + T - 1) / T, T, 0, stream>>>((const unsigned*)upd_flag,
                                                   (const unsigned*)occ_vol, out_upd);
}